// DotProductAttentionPooling_34471407518390
// MI455X (gfx1250) — compile-verified
//
#include <hip/hip_runtime.h>
#include <hip/hip_bf16.h>
#include <math.h>

// ---------------------------------------------------------------------------
// DotProductAttentionPooling for MI455X (gfx1250, wave32).
//   scores = (x . q) / 16, masked softmax over L, pooled = attn @ x
// Memory-bound: x = 256MB dominates (23.3 TB/s -> ~11us if read once).
// Strategy: flash-style chunked softmax, x streamed from HBM exactly once,
// staged into LDS by the Tensor Data Mover (one tensor_load_to_lds per block,
// with D# padding producing bank-conflict-free 260-float rows), scores via
// V_WMMA_F32_16X16X4_F32 (fp32 matrix path -> exact reference numerics),
// pooling partial accumulated from the same LDS tile.
// ---------------------------------------------------------------------------

typedef __attribute__((ext_vector_type(2))) float    v2f;
typedef __attribute__((ext_vector_type(4))) float    v4f;
typedef __attribute__((ext_vector_type(8))) float    v8f;
typedef __attribute__((ext_vector_type(4))) unsigned u32x4;
typedef __attribute__((ext_vector_type(8))) int      i32x8;
typedef __attribute__((ext_vector_type(4))) int      i32x4;

#define BB      32
#define LL      8192
#define DD      256
#define ROWS    128                 // rows per chunk (8 waves x 16-row WMMA tile)
#define CHUNKS  (LL / ROWS)         // 64
#define PAD     4
#define STRIDE  (DD + PAD)          // 260 floats: breaks 64-bank row aliasing
#define REC     (2 + DD)            // per-chunk record: m, s, acc[256]
#define NEG_INF (-__builtin_inff())

// LDS float layout (dynamic): tile[ROWS*STRIDE] | sc[ROWS] | red[2] | qs[DD]
#define OFF_SC   (ROWS * STRIDE)            // 33280
#define OFF_RED  (OFF_SC + ROWS)            // 33408
#define OFF_QS   (OFF_RED + 2)              // 33410 (even -> 8B aligned v2f)
#define SMEM_FLT (OFF_QS + DD)

__global__ __launch_bounds__(256) void attn_pool_chunk(
    const float* __restrict__ x, const int* __restrict__ mask,
    const float* __restrict__ query, float* __restrict__ ws)
{
    extern __shared__ float smem[];
    float* tile = smem;
    float* sc   = smem + OFF_SC;
    float* red  = smem + OFF_RED;
    float* qs   = smem + OFF_QS;

    const int t    = threadIdx.x;
    const int wave = t >> 5;
    const int lane = t & 31;
    const int half = lane >> 4;     // K sub-pair select (WMMA A/B layout)
    const int mrow = lane & 15;     // M row within 16-row tile

    const int blk = blockIdx.x;
    const int b   = blk / CHUNKS;
    const int c   = blk % CHUNKS;
    const int l0  = c * ROWS;

    const float* xc = x + ((size_t)b * LL + l0) * DD;

    if (t < DD) qs[t] = query[t];

    // ---- stage 128x256 fp32 tile into LDS via Tensor Data Mover -----------
    // D#: 2D tensor == tile (256 x 128 f32), LDS padding +4 dwords per 256
    // dwords -> row stride 260 floats. One DMA per block, issued by wave 0;
    // TENSORcnt tracks completion, barrier publishes LDS to all waves.
#if __has_builtin(__builtin_amdgcn_tensor_load_to_lds)
    if (wave == 0) {
        const unsigned long long ga = (unsigned long long)(uintptr_t)xc;
        u32x4 g0;
        g0[0] = 1u;                                   // count=1, user load D#
        g0[1] = 0u;                                   // lds_addr = 0 (dyn LDS base)
        g0[2] = (unsigned)(ga & 0xffffffffu);         // global_addr[31:0]
        g0[3] = (unsigned)((ga >> 32) & 0x01ffffffu)  // global_addr[56:32]
              | (2u << 30);                           // type = 2 ("image")
        i32x8 g1;
        g1[0] = (2 << 16)                             // data_size = 4B
              | (1 << 20)                             // pad_enable
              | (7 << 22)                             // pad_interval: 256 dwords
              | (3 << 25);                            // pad_amount:   4 dwords
        g1[1] = (int)((unsigned)DD << 16);            // tensor_dim0 = 256 (lo16)
        g1[2] = (int)((unsigned)ROWS << 16);          // dim0 hi=0 | tensor_dim1=128
        g1[3] = (int)((unsigned)DD << 16);            // dim1 hi=0 | tile_dim0 = 256
        g1[4] = ROWS;                                 // tile_dim1=128, tile_dim2=0
        g1[5] = DD;                                   // tensor_dim0_stride = 256
        g1[6] = (int)((((unsigned)(ROWS * DD)) & 0xffffu) << 16); // stride1 lo16
        g1[7] = (int)(((unsigned)(ROWS * DD)) >> 16);             // stride1 hi
        i32x4 z4 = {0, 0, 0, 0};                      // groups 2/3 unused (2D)
        i32x8 z8 = {0, 0, 0, 0, 0, 0, 0, 0};          // extra group (clang-23 ABI)
        __builtin_amdgcn_tensor_load_to_lds(g0, g1, z4, z4, z8, 0);
        __builtin_amdgcn_s_wait_tensorcnt(0);
    }
#else
    #pragma unroll
    for (int i = 0; i < (ROWS * DD / 4) / 256; ++i) {
        int f  = t + 256 * i;
        int r  = f >> 6;
        int c4 = f & 63;
        v4f v = *(const v4f*)(xc + r * DD + c4 * 4);
        *(v4f*)(tile + r * STRIDE + c4 * 4) = v;
    }
#endif
    __syncthreads();

    // ---- scores via chained V_WMMA_F32_16X16X4_F32 (16 rows per wave) -----
    // A: lane (mrow) holds row, {a.x,a.y} = x[row][k0+2*half .. +1] per ISA
    //    16x4 f32 A layout. B: query segment replicated into all 16 columns
    //    (same K striping as A), so every D column carries the 16 dots.
    {
        const float* arow = tile + (wave * 16 + mrow) * STRIDE;
        v8f acc = {0.f,0.f,0.f,0.f,0.f,0.f,0.f,0.f};
        #pragma unroll 4
        for (int k0 = 0; k0 < DD; k0 += 4) {
            v2f a  = *(const v2f*)(arow + k0 + 2 * half);
            v2f bv = *(const v2f*)(qs   + k0 + 2 * half);
            acc = __builtin_amdgcn_wmma_f32_16x16x4_f32(
                      false, a, false, bv, (short)0, acc, false, false);
        }
        // D layout: lane<16 VGPR r -> M=r, lane>=16 -> M=8+r (column N=lane&15,
        // all columns identical). Lanes 0 and 16 of each wave publish 8 rows.
        if (mrow == 0) {
            int rbase = wave * 16 + half * 8;
            #pragma unroll
            for (int r = 0; r < 8; ++r) {
                int row = rbase + r;
                int mk  = mask[(size_t)b * LL + l0 + row];
                sc[row] = mk ? acc[r] * 0.0625f : NEG_INF;   // 1/sqrt(256)
            }
        }
    }
    __syncthreads();

    // ---- chunk softmax partials: m = max, w = exp(s-m), s = sum(w) --------
    if (wave == 0) {
        float m0 = NEG_INF;
        #pragma unroll
        for (int i = 0; i < ROWS / 32; ++i)
            m0 = fmaxf(m0, sc[lane + 32 * i]);
        #pragma unroll
        for (int off = 16; off > 0; off >>= 1)
            m0 = fmaxf(m0, __shfl_xor(m0, off, 32));
        if (lane == 0) red[0] = m0;
    }
    __syncthreads();
    const float mval = red[0];

    if (t < ROWS) {
        float s = sc[t];
        sc[t] = (s == NEG_INF) ? 0.0f : expf(s - mval);   // all-masked safe
    }
    __syncthreads();

    if (wave == 0) {
        float s0 = 0.0f;
        #pragma unroll
        for (int i = 0; i < ROWS / 32; ++i)
            s0 += sc[lane + 32 * i];
        #pragma unroll
        for (int off = 16; off > 0; off >>= 1)
            s0 += __shfl_xor(s0, off, 32);
        if (lane == 0) red[1] = s0;
    }

    // ---- pooling partial: one thread per d, conflict-free LDS sweep -------
    float acc_d = 0.0f;
    #pragma unroll 8
    for (int l = 0; l < ROWS; ++l)
        acc_d = fmaf(sc[l], tile[l * STRIDE + t], acc_d);
    __syncthreads();

    float* rec = ws + (size_t)blk * REC;
    if (t == 0) rec[0] = mval;
    if (t == 1) rec[1] = red[1];
    rec[2 + t] = acc_d;
}

__global__ __launch_bounds__(256) void attn_pool_reduce(
    const float* __restrict__ ws, float* __restrict__ out)
{
    const int b = blockIdx.x;
    const int t = threadIdx.x;
    const float* base = ws + (size_t)b * CHUNKS * REC;

    float M = NEG_INF;
    #pragma unroll 4
    for (int c = 0; c < CHUNKS; ++c)
        M = fmaxf(M, base[(size_t)c * REC]);

    float S = 0.0f, acc = 0.0f;
    #pragma unroll 4
    for (int c = 0; c < CHUNKS; ++c) {
        float mc = base[(size_t)c * REC];
        float f  = (mc == NEG_INF) ? 0.0f : expf(mc - M);
        S   += base[(size_t)c * REC + 1] * f;
        acc += base[(size_t)c * REC + 2 + t] * f;
    }
    // fully-masked batch: softmax is NaN -> nan_to_num -> 0
    out[(size_t)b * DD + t] = (S > 0.0f) ? (acc / S) : 0.0f;
}

extern "C" void kernel_launch(void* const* d_in, const int* in_sizes, int n_in,
                              void* d_out, int out_size, void* d_ws, size_t ws_size,
                              hipStream_t stream) {
    const float* x     = (const float*)d_in[0];
    const int*   mask  = (const int*)d_in[1];
    const float* query = (const float*)d_in[2];
    float* out = (float*)d_out;
    float* ws  = (float*)d_ws;      // needs BB*CHUNKS*REC*4 = ~2.1 MB

    size_t smem = (size_t)SMEM_FLT * sizeof(float);   // ~134.7 KB (<320KB/WGP)
    attn_pool_chunk<<<BB * CHUNKS, 256, smem, stream>>>(x, mask, query, ws);
    attn_pool_reduce<<<BB, 256, 0, stream>>>(ws, out);
}